// MultiProbabilisticSkipper_7773890806434
// MI455X (gfx1250) — compile-verified
//
#include <hip/hip_runtime.h>

#define NB 8
#define NS 4096
#define ND 1024

// 128-bit payload type the gfx1250 async builtins expect.
typedef int v4i_t __attribute__((vector_size(16)));
typedef __attribute__((address_space(1))) v4i_t* gv4_t;  // global (AS1)
typedef __attribute__((address_space(3))) v4i_t* lv4_t;  // LDS (AS3)

// ---------------------------------------------------------------------------
// CDNA5 async global<->LDS helpers (ASYNCcnt path). Builtins confirmed present
// on this toolchain; inline-asm fallback kept for portability.
// ---------------------------------------------------------------------------
#if __has_builtin(__builtin_amdgcn_global_load_async_to_lds_b128)
#define HAVE_ASYNC_LDS_BUILTINS 1
#endif

__device__ __forceinline__ void async_load_b128(const float* g, float* l) {
#if defined(HAVE_ASYNC_LDS_BUILTINS)
  __builtin_amdgcn_global_load_async_to_lds_b128(
      (gv4_t)g, (lv4_t)l, /*offset=*/0, /*cpol=*/0);
#else
  unsigned lds_addr =
      (unsigned)(unsigned long)(__attribute__((address_space(3))) void*)l;
  asm volatile("global_load_async_to_lds_b128 %0, %1, off"
               :: "v"(lds_addr), "v"(g)
               : "memory");
#endif
}

__device__ __forceinline__ void async_store_b128(float* g, const float* l) {
#if defined(HAVE_ASYNC_LDS_BUILTINS)
  __builtin_amdgcn_global_store_async_from_lds_b128(
      (gv4_t)g, (lv4_t)l, /*offset=*/0, /*cpol=*/0);
#else
  unsigned lds_addr =
      (unsigned)(unsigned long)(__attribute__((address_space(3))) void*)l;
  asm volatile("global_store_async_from_lds_b128 %0, %1, off"
               :: "v"(g), "v"(lds_addr)
               : "memory");
#endif
}

__device__ __forceinline__ void wait_asynccnt0() {
#if __has_builtin(__builtin_amdgcn_s_wait_asynccnt)
  __builtin_amdgcn_s_wait_asynccnt(0);
#else
  asm volatile("s_wait_asynccnt 0" ::: "memory");
#endif
}

// ---------------------------------------------------------------------------
// Kernel (defined first so it leads the .text section / disasm snippet):
// one 256-thread block per output row (b, p). Kept rows are moved with the
// CDNA5 async DMA path: global -> LDS -> global, one B128 per lane per tensor
// (256 lanes * 16B = 4KB = one D-row). Tail rows are zero-filled.
// Branch on (p < n_kept[b]) is block-uniform, so EXEC is all-ones through the
// async ops. Per-lane LDS load/store regions are wave-disjoint, so a single
// wave-level s_wait_asynccnt 0 orders the LDS producer/consumer; s_endpgm's
// implicit wait-idle covers the outstanding async stores.
// ---------------------------------------------------------------------------
__global__ __launch_bounds__(256) void compact_rows_kernel(
    const float* __restrict__ x0, const float* __restrict__ x1,
    float* __restrict__ out0, float* __restrict__ out1,
    const int* __restrict__ src_idx, const int* __restrict__ n_kept) {
  const int p = blockIdx.x;   // output row within batch
  const int b = blockIdx.y;   // batch
  const int t = threadIdx.x;  // 0..255

  __shared__ float lbuf[2 * ND];  // 8KB: row of x0, row of x1

  const int nk = n_kept[b];
  const size_t row_out = ((size_t)b * NS + p) * ND;

  if (p < nk) {
    const int s = src_idx[(size_t)b * NS + p];
    const size_t row_in = ((size_t)b * NS + s) * ND;
    const int e = t * 4;  // 4 floats (16B) per lane per tensor

    async_load_b128(x0 + row_in + e, &lbuf[e]);
    async_load_b128(x1 + row_in + e, &lbuf[ND + e]);
    wait_asynccnt0();  // LDS now holds both rows (wave-local producer/consumer)
    async_store_b128(out0 + row_out + e, &lbuf[e]);
    async_store_b128(out1 + row_out + e, &lbuf[ND + e]);
  } else {
    const float4 z = make_float4(0.f, 0.f, 0.f, 0.f);
    *(float4*)(out0 + row_out + t * 4) = z;
    *(float4*)(out1 + row_out + t * 4) = z;
  }
}

// ---------------------------------------------------------------------------
// Per-batch stable scan of keep[] -> src_idx (scatter) + n_kept.
// One block per batch; 1024 threads x 4 keep values = S.
// ---------------------------------------------------------------------------
__global__ __launch_bounds__(1024) void skip_scan_kernel(
    const int* __restrict__ keep, int* __restrict__ src_idx,
    int* __restrict__ n_kept) {
  const int b = blockIdx.x;
  const int t = threadIdx.x;
  __shared__ int sums[1024];

  const int* krow = keep + (size_t)b * NS;
  const int base = t * 4;
  int k[4];
  int local = 0;
#pragma unroll
  for (int i = 0; i < 4; ++i) {
    k[i] = krow[base + i];
    local += k[i];
  }
  sums[t] = local;
  __syncthreads();

  // Hillis-Steele inclusive scan over 1024 partials.
  for (int off = 1; off < 1024; off <<= 1) {
    int v = (t >= off) ? sums[t - off] : 0;
    __syncthreads();
    sums[t] += v;
    __syncthreads();
  }

  const int incl = sums[t];
  int pos = incl - local;  // exclusive prefix
#pragma unroll
  for (int i = 0; i < 4; ++i) {
    if (k[i]) {
      src_idx[(size_t)b * NS + pos] = base + i;
      ++pos;
    }
  }
  if (t == 1023) n_kept[b] = incl;
}

// ---------------------------------------------------------------------------
// Launch. Inputs: [0]=x0 (f32 B*S*D), [1]=x1 (f32 B*S*D),
// [2]=skipping_probs (unused), [3]=keep (i32 B*S).
// Output: out0 ++ out1 (f32, 2*B*S*D).
// Workspace: int src_idx[B*S], int n_kept[B]  (~128KB).
// ---------------------------------------------------------------------------
extern "C" void kernel_launch(void* const* d_in, const int* in_sizes, int n_in,
                              void* d_out, int out_size, void* d_ws,
                              size_t ws_size, hipStream_t stream) {
  (void)in_sizes; (void)n_in; (void)out_size; (void)ws_size;

  const float* x0 = (const float*)d_in[0];
  const float* x1 = (const float*)d_in[1];
  const int* keep = (const int*)d_in[3];

  float* out0 = (float*)d_out;
  float* out1 = out0 + (size_t)NB * NS * ND;

  int* src_idx = (int*)d_ws;
  int* n_kept = src_idx + (size_t)NB * NS;

  skip_scan_kernel<<<NB, 1024, 0, stream>>>(keep, src_idx, n_kept);
  compact_rows_kernel<<<dim3(NS, NB), 256, 0, stream>>>(x0, x1, out0, out1,
                                                        src_idx, n_kept);
}